// SpikeFP64Adder_29875792511511
// MI455X (gfx1250) — compile-verified
//
#include <hip/hip_runtime.h>
#include <stdint.h>

// ---------------------------------------------------------------------------
// Bit-faithful FP64 "spike" adder, transcribed from the reference. All
// arithmetic is exact uint64/uint32 bit manipulation (57-bit internal
// mantissa, 11-bit wrapped exponents, faithful round-carry-at-bit-53 quirk).
// ---------------------------------------------------------------------------
__device__ __forceinline__ uint64_t fp64_spike_add(uint64_t A, uint64_t B) {
    const uint64_t MASK52 = (1ull << 52) - 1;
    const uint64_t MASK57 = (1ull << 57) - 1;

    uint32_t s_a = (uint32_t)(A >> 63);
    uint32_t s_b = (uint32_t)(B >> 63);
    uint32_t e_a = (uint32_t)(A >> 52) & 0x7FFu;
    uint32_t e_b = (uint32_t)(B >> 52) & 0x7FFu;
    uint64_t m_a_raw = A & MASK52;
    uint64_t m_b_raw = B & MASK52;

    uint64_t hid_a = (e_a != 0) ? 1ull : 0ull;
    uint64_t hid_b = (e_b != 0) ? 1ull : 0ull;
    uint32_t e_a_eff = e_a ? e_a : 1u;
    uint32_t e_b_eff = e_b ? e_b : 1u;

    uint64_t m_a   = (hid_a << 56) | (m_a_raw << 4);   // 57-bit internal
    uint64_t m_b   = (hid_b << 56) | (m_b_raw << 4);
    uint64_t m_a53 = (hid_a << 52) | m_a_raw;
    uint64_t m_b53 = (hid_b << 52) | m_b_raw;

    bool a_abs_eq_b = (e_a_eff == e_b_eff) && (m_a53 == m_b53);
    bool a_ge_b     = (e_a_eff > e_b_eff) ||
                      ((e_a_eff == e_b_eff) && (m_a53 >= m_b53));

    uint32_t exp_diff = a_ge_b ? (e_a_eff - e_b_eff) : (e_b_eff - e_a_eff);
    bool is_big_diff  = ((exp_diff >> 6) != 0u) || (((exp_diff >> 3) & 7u) == 7u);
    uint32_t shift_amt = exp_diff & 63u;

    uint32_t e_max     = a_ge_b ? e_a_eff : e_b_eff;
    uint64_t m_large   = a_ge_b ? m_a : m_b;
    uint64_t m_small_u = a_ge_b ? m_b : m_a;
    uint64_t m_small   = (m_small_u >> shift_amt) & MASK57;
    bool shift_sticky  = (m_small_u & ((1ull << shift_amt) - 1ull)) != 0ull;

    bool is_diff_sign = (s_a != s_b);
    bool exact_cancel = is_diff_sign && a_abs_eq_b;
    uint32_t s_large  = a_ge_b ? s_a : s_b;

    uint64_t sum57      = m_large + m_small;
    uint32_t sum_carry  = (uint32_t)((sum57 >> 57) & 1ull);
    uint64_t sum_result = sum57 & MASK57;
    uint64_t diff_result = (m_large - m_small) & MASK57;
    bool need_sub_one   = is_diff_sign && shift_sticky;
    uint64_t diff_final = need_sub_one ? ((diff_result - 1ull) & MASK57) : diff_result;
    uint64_t mant       = is_diff_sign ? diff_final : sum_result;
    uint32_t result_carry = is_diff_sign ? 0u : sum_carry;

    // leading-zero count over the 57-bit field (57 when mant == 0)
    uint32_t lzc = mant ? ((uint32_t)__clzll((long long)mant) - 7u) : 57u;

    bool is_underflow   = (lzc >= e_max);
    uint64_t norm_mant  = (mant << lzc) & MASK57;
    uint32_t e_after_norm = (e_max - lzc) & 0x7FFu;   // wraparound faithful
    uint32_t e_plus_one   = (e_max + 1u) & 0x7FFu;
    uint32_t e_normal     = is_underflow ? 0u : e_after_norm;
    bool rc = (result_carry == 1u);
    uint32_t final_e_pre  = rc ? e_plus_one : e_normal;

    uint64_t m_overflow     = mant >> 5;
    uint32_t round_overflow = (uint32_t)((mant >> 4) & 1ull);
    bool sticky_overflow    = (mant & 0xFull) != 0ull;
    uint64_t m_norm         = (norm_mant >> 4) & MASK52;
    uint32_t round_norm     = (uint32_t)((norm_mant >> 3) & 1ull);
    bool sticky_norm        = (norm_mant & 7ull) != 0ull;
    uint64_t m_subnorm      = mant >> 5;

    uint64_t m_pre     = rc ? m_overflow : m_norm;
    uint32_t round_pre = rc ? round_overflow : round_norm;
    bool sticky_pre    = (rc ? sticky_overflow : sticky_norm) ||
                         ((!is_diff_sign) && shift_sticky);
    uint64_t m_sel = is_underflow ? m_subnorm : m_pre;
    uint32_t L = (uint32_t)(m_sel & 1ull);
    bool do_round = (round_pre == 1u) && (sticky_pre || (L == 1u)) && (!is_underflow);

    // faithful: carry-out is bit 53 of the 53-bit ripple sum
    uint64_t sum53 = m_sel + (do_round ? 1ull : 0ull);
    uint32_t round_carry = (uint32_t)((sum53 >> 53) & 1ull);
    uint64_t m_rounded = sum53 & MASK52;
    uint64_t m_final   = round_carry ? 0ull : m_rounded;
    uint32_t e_rounded = (final_e_pre + round_carry) & 0x7FFu;
    uint32_t computed_e = round_carry ? e_rounded : final_e_pre;
    uint32_t computed_s = s_large;

    uint32_t cs = exact_cancel ? 0u : computed_s;
    uint32_t ce = exact_cancel ? 0u : computed_e;
    uint64_t cm = exact_cancel ? 0ull : m_final;
    bool all_one = (computed_e == 0x7FFu);
    uint32_t fs = all_one ? computed_s : cs;
    uint32_t fe = all_one ? 0x7FFu : ce;
    uint64_t fm = all_one ? 0ull : cm;

    uint64_t res = ((uint64_t)fs << 63) | ((uint64_t)fe << 52) | fm;
    uint64_t larger = a_ge_b ? A : B;
    return is_big_diff ? larger : res;
}

// ---------------------------------------------------------------------------
// Tiled kernel: 64 threads / 64 elements per block.
// LDS layout: padded rows of 66 dwords (264 B) -> conflict-free b64 access
// (lane t hits banks {2t, 2t+1}). Global side of the async copies is fully
// coalesced (32 lanes x 8B contiguous per instruction). All global traffic
// is marked non-temporal: the 402 MB stream has zero reuse and exceeds the
// 192 MB L2, so NT keeps the stream from thrashing the cache.
// ---------------------------------------------------------------------------
#define ROWS      64
#define ROW_DW    66
#define ROW_BYTES 264
#define TILE_DW   (ROWS * ROW_DW)   // 4224 dwords = 16896 B per tile

__global__ __launch_bounds__(64)
void spike_fp64_add_kernel(const float* __restrict__ A,
                           const float* __restrict__ B,
                           float* __restrict__ Out,
                           long long nNum) {
    __shared__ uint32_t smem[2 * TILE_DW];   // A-tile | B-tile; out reuses A-tile

    const int t = threadIdx.x;
    const long long n0 = (long long)blockIdx.x * ROWS;
    const long long rowsLeft = nNum - n0;

    if (rowsLeft >= ROWS) {
        const unsigned ldsBase = (unsigned)(size_t)(&smem[0]);
        const uint8_t* aBase = (const uint8_t*)(A + n0 * 64);
        const uint8_t* bBase = (const uint8_t*)(B + n0 * 64);

        // Stage A and B tiles: 2048 8-byte chunks per tile, 64 threads.
        // Global: coalesced, non-temporal. LDS: same row/column layout, padded rows.
        for (int i = 0; i < 32; ++i) {
            unsigned c   = (unsigned)(i * 64 + t);
            unsigned row = c >> 5;
            unsigned col = c & 31u;
            unsigned goff = c * 8u;
            unsigned lA = ldsBase + row * ROW_BYTES + col * 8u;
            unsigned lB = lA + TILE_DW * 4u;
            asm volatile("global_load_async_to_lds_b64 %0, %1, %2 th:TH_LOAD_NT"
                         :: "v"(lA), "v"(goff), "s"(aBase) : "memory");
            asm volatile("global_load_async_to_lds_b64 %0, %1, %2 th:TH_LOAD_NT"
                         :: "v"(lB), "v"(goff), "s"(bBase) : "memory");
        }
        asm volatile("s_wait_asynccnt 0x0" ::: "memory");
        __syncthreads();

        // Pack this thread's two 64-bit operands from its padded LDS rows.
        const uint32_t* rowA = &smem[t * ROW_DW];
        const uint32_t* rowB = rowA + TILE_DW;
        uint64_t ua = 0, ub = 0;
#pragma unroll
        for (int k = 0; k < 32; ++k) {
            uint64_t wa = *reinterpret_cast<const uint64_t*>(rowA + 2 * k);
            uint64_t wb = *reinterpret_cast<const uint64_t*>(rowB + 2 * k);
            // float 1.0f = 0x3F800000 -> bit 23 of each dword is the bit value
            ua |= ((wa >> 23) & 1ull) << (63 - 2 * k);
            ua |= ((wa >> 55) & 1ull) << (62 - 2 * k);
            ub |= ((wb >> 23) & 1ull) << (63 - 2 * k);
            ub |= ((wb >> 55) & 1ull) << (62 - 2 * k);
        }

        uint64_t r = fp64_spike_add(ua, ub);

        // Unpack into LDS (reuse A-tile region: each thread owns its row).
        uint32_t* orow = &smem[t * ROW_DW];
#pragma unroll
        for (int k = 0; k < 32; ++k) {
            uint64_t lo = ((r >> (63 - 2 * k)) & 1ull) ? 0x3F800000ull : 0ull;
            uint64_t hi = ((r >> (62 - 2 * k)) & 1ull) ? 0x3F800000ull : 0ull;
            *reinterpret_cast<uint64_t*>(orow + 2 * k) = lo | (hi << 32);
        }
        __syncthreads();

        // Coalesced, non-temporal async store from LDS back to global.
        const uint8_t* oBase = (const uint8_t*)(Out + n0 * 64);
        for (int i = 0; i < 32; ++i) {
            unsigned c   = (unsigned)(i * 64 + t);
            unsigned row = c >> 5;
            unsigned col = c & 31u;
            unsigned goff = c * 8u;
            unsigned lO = ldsBase + row * ROW_BYTES + col * 8u;
            asm volatile("global_store_async_from_lds_b64 %0, %1, %2 th:TH_STORE_NT"
                         :: "v"(goff), "v"(lO), "s"(oBase) : "memory");
        }
        asm volatile("s_wait_asynccnt 0x0" ::: "memory");
    } else {
        // Tail block: plain scalar path (not hit for the reference sizes).
        long long n = n0 + t;
        if (n < nNum) {
            const float* a = A + n * 64;
            const float* b = B + n * 64;
            uint64_t ua = 0, ub = 0;
            for (int k = 0; k < 64; ++k) {
                ua |= (uint64_t)(a[k] != 0.0f) << (63 - k);
                ub |= (uint64_t)(b[k] != 0.0f) << (63 - k);
            }
            uint64_t r = fp64_spike_add(ua, ub);
            float* o = Out + n * 64;
            for (int k = 0; k < 64; ++k)
                o[k] = (float)((r >> (63 - k)) & 1ull);
        }
    }
}

extern "C" void kernel_launch(void* const* d_in, const int* in_sizes, int n_in,
                              void* d_out, int out_size, void* d_ws, size_t ws_size,
                              hipStream_t stream) {
    const float* A = (const float*)d_in[0];
    const float* B = (const float*)d_in[1];
    float* Out = (float*)d_out;

    long long nFloats = (long long)in_sizes[0];
    long long nNum = nFloats / 64;           // 524288 for the reference shapes
    if (nNum <= 0) return;

    long long blocks = (nNum + ROWS - 1) / ROWS;
    spike_fp64_add_kernel<<<dim3((unsigned)blocks), dim3(64), 0, stream>>>(A, B, Out, nNum);
}